// SectorGraphAttention_26843545600094
// MI455X (gfx1250) — compile-verified
//
#include <hip/hip_runtime.h>
#include <hip/hip_bf16.h>
#include <math.h>

// ---------------------------------------------------------------------------
// Fused sector-graph attention block for MI455X (gfx1250, wave32).
//   y = LN( x + (softmax(QK^T/sqrt(hd) + adj) V) Wo + bo )
// B=16384, S=12, D=512, H=8, hd=64.  Memory-bound (~0.81 GB min HBM traffic
// ~= 35us at 23.3 TB/s); fully fused so q/k/v/attn-out never touch HBM.
// GEMMs via V_WMMA_F32_16X16X4_F32 (fp32, compute is ~1% of the roofline).
// Weight staging is double-buffered through LDS with CDNA5 async
// global->LDS copies (ASYNCcnt) so L2 weight streams overlap the WMMAs
// (one WG per WGP at this LDS footprint => pipelining is the only overlap).
// ---------------------------------------------------------------------------

typedef __attribute__((ext_vector_type(2))) float v2f;
typedef __attribute__((ext_vector_type(8))) float v8f;
typedef int v4i_t __attribute__((vector_size(16)));   // matches async-LDS builtin

#define B_TOT 16384
#define S     12
#define D     512
#define H     8
#define HD    64
#define BB    4              // batches per workgroup
#define ROWS  48             // BB*S, == 3 M-tiles of 16 exactly
#define YSTR  516            // ys/outs col stride (bank-conflict padding)
#define WSTR  72             // QKV weight-stage col stride
#define QSTR  68             // qkvh col stride
#define W2STR 136            // out-proj weight-stage col stride
#define SSTR  20             // score tile col stride
#define QKV_CP 6             // async float4 copies per thread per QKV chunk
#define O_CP   4             // async float4 copies per thread per Wo chunk

static __device__ __forceinline__ v8f wmma_f32(v2f a, v2f b, v8f c) {
  // D = A(16x4 f32) * B(4x16 f32) + C(16x16 f32)
  return __builtin_amdgcn_wmma_f32_16x16x4_f32(false, a, false, b,
                                               (short)0, c, false, false);
}

// 16-byte global -> LDS copy. Async (ASYNCcnt-tracked, no VGPR data) when the
// CDNA5 builtin exists; falls back to load+ds_store otherwise.
static __device__ __forceinline__ void cp16(void* lds_dst, const void* gsrc) {
#if defined(__gfx1250__) && __has_builtin(__builtin_amdgcn_global_load_async_to_lds_b128)
  __builtin_amdgcn_global_load_async_to_lds_b128(
      (__attribute__((address_space(1))) v4i_t*)gsrc,
      (__attribute__((address_space(3))) v4i_t*)lds_dst, 0, 0);
#else
  *(float4*)lds_dst = *(const float4*)gsrc;
#endif
}

template <int N>
static __device__ __forceinline__ void wait_async() {
#if defined(__gfx1250__)
#if __has_builtin(__builtin_amdgcn_s_wait_asynccnt)
  __builtin_amdgcn_s_wait_asynccnt(N);
#else
  asm volatile("s_wait_asynccnt %0" ::"i"(N) : "memory");
#endif
#endif
}

__global__ __launch_bounds__(256, 1)
void fused_sga_kernel(const float* __restrict__ x,
                      const float* __restrict__ Wq, const float* __restrict__ bq,
                      const float* __restrict__ Wk, const float* __restrict__ bk,
                      const float* __restrict__ Wv, const float* __restrict__ bv,
                      const float* __restrict__ adj,
                      const float* __restrict__ Wo, const float* __restrict__ bo,
                      const float* __restrict__ gamma, const float* __restrict__ beta,
                      float* __restrict__ out) {
  __shared__ float ys[ROWS * YSTR];          // x block, later y (resid) block
  __shared__ float outs[ROWS * YSTR];        // attention output (48 x 512)
  __shared__ float wst[2 * 3 * 32 * WSTR];   // double-buffered weight chunks
  __shared__ float qkvh[3 * ROWS * QSTR];    // per-head q,k,v (48 x 64 each)
  __shared__ float sc[BB * 16 * SSTR];       // per-batch softmaxed score tiles

  const int tid  = threadIdx.x;
  const int lane = tid & 31;
  const int wave = tid >> 5;
  const int hi   = lane >> 4;       // 0: lanes 0-15, 1: lanes 16-31
  const int l15  = lane & 15;
  const int row0 = blockIdx.x * ROWS;                 // global GEMM row base
  const float* xg = x + (size_t)row0 * D;

  const float* Wqkv[3] = {Wq, Wk, Wv};
  const float* bqkv[3] = {bq, bk, bv};

  // QKV weight chunk (Kc=32, one head's 64 cols, 3 matrices): 1536 float4.
  // Issued asynchronously; completion published by wait_async + barrier.
  auto stage_qkv = [&](int h, int kc, float* buf) {
    #pragma unroll
    for (int it = 0; it < QKV_CP; ++it) {
      int idx = it * 256 + tid;
      int mat = idx >> 9;
      int rem = idx & 511;
      int kk  = rem >> 4;                    // 0..31
      int c   = (rem & 15) << 2;
      cp16(&buf[(mat * 32 + kk) * WSTR + c],
           Wqkv[mat] + (size_t)(kc * 32 + kk) * D + h * HD + c);
    }
  };
  // Wo chunk (Kc=32 x Nblk=128): 1024 float4.
  auto stage_o = [&](int nblk, int kc, float* buf) {
    #pragma unroll
    for (int it = 0; it < O_CP; ++it) {
      int idx = it * 256 + tid;
      int kk  = idx >> 5;                    // 0..31
      int c   = (idx & 31) << 2;
      cp16(&buf[kk * W2STR + c],
           Wo + (size_t)(kc * 32 + kk) * D + nblk * 128 + c);
    }
  };

  // ---- prefetch first QKV weight chunk, then stage x block (48x512) -------
  stage_qkv(0, 0, &wst[0]);
  #pragma unroll
  for (int it = 0; it < 24; ++it) {          // 6144 float4 total
    int idx = it * 256 + tid;
    int row = idx >> 7;                      // 128 float4 per row
    int c   = (idx & 127) << 2;
    cp16(&ys[row * YSTR + c], xg + row * D + c);
  }
  wait_async<0>();
  __syncthreads();

  // ---------------- per-head: QKV projection + attention --------------------
  for (int h = 0; h < H; ++h) {
    // 36 output tiles: 3 mats x 3 M-tiles x 4 N-tiles, round-robin over waves
    v8f acc[5] = {};
    int tmat[5], tmt[5], tnt[5];
    int ntile = 0;
    for (int t = wave; t < 36; t += 8) {
      tmat[ntile] = t / 12;
      tmt[ntile]  = (t % 12) >> 2;
      tnt[ntile]  = t & 3;
      ++ntile;
    }

    for (int kc = 0; kc < 16; ++kc) {        // K chunks of 32, pipelined
      float* cur = &wst[(kc & 1) * 3 * 32 * WSTR];
      if (kc + 1 < 16) {                     // prefetch next chunk
        stage_qkv(h, kc + 1, &wst[((kc + 1) & 1) * 3 * 32 * WSTR]);
        wait_async<QKV_CP>();                // drain current chunk (in-order)
      } else {
        wait_async<0>();
      }
      __syncthreads();

      #pragma unroll
      for (int i = 0; i < 5; ++i) {
        if (i < ntile) {
          const float* A  = &ys[(tmt[i] * 16) * YSTR + kc * 32];
          const float* Bm = &cur[tmat[i] * 32 * WSTR + tnt[i] * 16];
          #pragma unroll
          for (int k4 = 0; k4 < 8; ++k4) {
            int kk = (k4 << 2) + (hi << 1);
            v2f a = *(const v2f*)(&A[l15 * YSTR + kk]);      // A[M=l15][kk,kk+1]
            v2f b;                                           // B[kk..][N=l15]
            b.x = Bm[kk * WSTR + l15];
            b.y = Bm[(kk + 1) * WSTR + l15];
            acc[i] = wmma_f32(a, b, acc[i]);
          }
        }
      }
      __syncthreads();                       // chunk buffer free for reuse
    }

    // store q,k,v head block to LDS (+bias, q pre-scaled by 1/sqrt(hd))
    #pragma unroll
    for (int i = 0; i < 5; ++i) {
      if (i < ntile) {
        int   mat   = tmat[i];
        float bias  = bqkv[mat][h * HD + tnt[i] * 16 + l15];
        float scale = (mat == 0) ? 0.125f : 1.0f;
        #pragma unroll
        for (int r = 0; r < 8; ++r) {
          int row = tmt[i] * 16 + r + hi * 8;
          qkvh[(mat * ROWS + row) * QSTR + tnt[i] * 16 + l15] =
              (acc[i][r] + bias) * scale;
        }
      }
    }
    __syncthreads();

    // prefetch chunk 0 of the next phase so it streams under the attention
    if (h + 1 < H) stage_qkv(h + 1, 0, &wst[0]);
    else           stage_o(0, 0, &wst[0]);

    // ---- attention for this head: wave bb handles batch bb (waves 0-3) ----
    if (wave < BB) {
      const int bb    = wave;
      const int rbase = bb * S;
      const float* Q = &qkvh[0 * ROWS * QSTR];
      const float* K = &qkvh[1 * ROWS * QSTR];
      const float* V = &qkvh[2 * ROWS * QSTR];

      // scores (16x16, 12x12 valid) = q @ k^T  (K^T via B read pattern)
      v8f cs = {};
      #pragma unroll
      for (int k4 = 0; k4 < 16; ++k4) {
        int kk   = (k4 << 2) + (hi << 1);
        int arow = rbase + l15; arow = arow < ROWS ? arow : ROWS - 1;
        v2f a = *(const v2f*)(&Q[arow * QSTR + kk]);
        v2f b;                                // B[k=kk][n] = K[row n][kk]
        b.x = K[arow * QSTR + kk];
        b.y = K[arow * QSTR + kk + 1];
        cs = wmma_f32(a, b, cs);
      }

      // + adjacency bias, masked softmax over t<12, write attn tile to LDS
      #pragma unroll
      for (int r = 0; r < 8; ++r) {
        int s_ = r + hi * 8;
        float val = cs[r];
        if (s_ < S && l15 < S) val += adj[(h * S + s_) * S + l15];
        val = (l15 < S) ? val : -3.0e38f;
        float m = val;
        #pragma unroll
        for (int o = 1; o < 16; o <<= 1) m = fmaxf(m, __shfl_xor(m, o, 32));
        float p = __expf(val - m);            // masked cols -> 0
        float ssum = p;
        #pragma unroll
        for (int o = 1; o < 16; o <<= 1) ssum += __shfl_xor(ssum, o, 32);
        sc[(bb * 16 + s_) * SSTR + l15] = p / ssum;
      }

      // out_h = attn(16x16, K-pad zero) @ v(12x64)
      #pragma unroll
      for (int nt = 0; nt < 4; ++nt) {
        v8f co = {};
        #pragma unroll
        for (int k4 = 0; k4 < 4; ++k4) {
          int kk = (k4 << 2) + (hi << 1);
          v2f a = *(const v2f*)(&sc[(bb * 16 + l15) * SSTR + kk]);
          int br0 = rbase + kk;     br0 = br0 < ROWS ? br0 : ROWS - 1;
          int br1 = rbase + kk + 1; br1 = br1 < ROWS ? br1 : ROWS - 1;
          v2f b;
          b.x = V[br0 * QSTR + nt * 16 + l15];
          b.y = V[br1 * QSTR + nt * 16 + l15];
          co = wmma_f32(a, b, co);
        }
        #pragma unroll
        for (int r = 0; r < 8; ++r) {
          int lr = r + hi * 8;
          if (lr < S)
            outs[(rbase + lr) * YSTR + h * HD + nt * 16 + l15] = co[r];
        }
      }
    }
    __syncthreads();
  }

  // ---------------- output projection + bias + residual ---------------------
  // (chunk (0,0) already streaming; pipeline over flattened (nblk,kc))
  for (int nblk = 0; nblk < 4; ++nblk) {     // N blocks of 128
    v8f acc2[3] = {};
    for (int kc = 0; kc < 16; ++kc) {        // K chunks of 32
      int cg = nblk * 16 + kc;
      float* cur = &wst[(cg & 1) * 32 * W2STR];
      if (cg + 1 < 64) {
        stage_o((cg + 1) >> 4, (cg + 1) & 15, &wst[((cg + 1) & 1) * 32 * W2STR]);
        wait_async<O_CP>();
      } else {
        wait_async<0>();
      }
      __syncthreads();

      #pragma unroll
      for (int i = 0; i < 3; ++i) {          // mt = i, nt = wave
        const float* A  = &outs[(i * 16) * YSTR + kc * 32];
        const float* Bm = &cur[wave * 16];
        #pragma unroll
        for (int k4 = 0; k4 < 8; ++k4) {
          int kk = (k4 << 2) + (hi << 1);
          v2f a = *(const v2f*)(&A[l15 * YSTR + kk]);
          v2f b;
          b.x = Bm[kk * W2STR + l15];
          b.y = Bm[(kk + 1) * W2STR + l15];
          acc2[i] = wmma_f32(a, b, acc2[i]);
        }
      }
      __syncthreads();
    }
    // y = proj + bo + x  (overwrite ys in place; each (row,col) hit once)
    #pragma unroll
    for (int i = 0; i < 3; ++i) {
      int   col  = nblk * 128 + wave * 16 + l15;
      float bias = bo[col];
      #pragma unroll
      for (int r = 0; r < 8; ++r) {
        int row = i * 16 + r + hi * 8;
        ys[row * YSTR + col] = acc2[i][r] + bias + ys[row * YSTR + col];
      }
    }
  }
  __syncthreads();

  // ---------------- LayerNorm + store ---------------------------------------
  for (int rr = wave; rr < ROWS; rr += 8) {
    const float* yr = &ys[rr * YSTR];
    float s1 = 0.f;
    #pragma unroll
    for (int j = 0; j < 16; ++j) s1 += yr[j * 32 + lane];
    #pragma unroll
    for (int o = 1; o < 32; o <<= 1) s1 += __shfl_xor(s1, o, 32);
    float mu = s1 * (1.0f / 512.0f);

    float s2 = 0.f;
    #pragma unroll
    for (int j = 0; j < 16; ++j) {
      float d = yr[j * 32 + lane] - mu;
      s2 += d * d;
    }
    #pragma unroll
    for (int o = 1; o < 32; o <<= 1) s2 += __shfl_xor(s2, o, 32);
    float rs = rsqrtf(s2 * (1.0f / 512.0f) + 1e-5f);

    float* og = out + (size_t)(row0 + rr) * D;
    #pragma unroll
    for (int j = 0; j < 16; ++j) {
      int c = j * 32 + lane;
      og[c] = (yr[c] - mu) * rs * gamma[c] + beta[c];
    }
  }
}

extern "C" void kernel_launch(void* const* d_in, const int* in_sizes, int n_in,
                              void* d_out, int out_size, void* d_ws, size_t ws_size,
                              hipStream_t stream) {
  (void)in_sizes; (void)n_in; (void)out_size; (void)d_ws; (void)ws_size;
  const float* x     = (const float*)d_in[0];
  const float* Wq    = (const float*)d_in[1];
  const float* bq    = (const float*)d_in[2];
  const float* Wk    = (const float*)d_in[3];
  const float* bk    = (const float*)d_in[4];
  const float* Wv    = (const float*)d_in[5];
  const float* bv    = (const float*)d_in[6];
  const float* adj   = (const float*)d_in[7];
  const float* Wo    = (const float*)d_in[8];
  const float* bo    = (const float*)d_in[9];
  const float* gamma = (const float*)d_in[10];
  const float* beta  = (const float*)d_in[11];
  float* out = (float*)d_out;

  dim3 grid(B_TOT / BB);   // 4096 workgroups, 4 batches each
  dim3 block(256);         // 8 wave32 waves
  fused_sga_kernel<<<grid, block, 0, stream>>>(x, Wq, bq, Wk, bk, Wv, bv,
                                               adj, Wo, bo, gamma, beta, out);
}